// WormholeAttention_56762287784273
// MI455X (gfx1250) — compile-verified
//
#include <hip/hip_runtime.h>
#include <math.h>

typedef __attribute__((ext_vector_type(2))) float v2f;
typedef __attribute__((ext_vector_type(8))) float v8f;

constexpr int HWn  = 2304;      // 48*48
constexpr int Tn   = 32;
constexpr int THWn = 73728;     // T*HW
constexpr int Dn   = 256;
constexpr int An   = 128;
constexpr int Kn   = 16;
constexpr int QROWS = 64;       // query rows per block in sim kernel
constexpr int NWAVE = 8;
constexpr int KEYTILES = THWn / 16;            // 4608
constexpr int KEYSPLIT = 8;                    // key-dim blocks
constexpr int TILES_PER_SPLIT = KEYTILES / KEYSPLIT;  // 576
constexpr int QGROUPS = HWn / QROWS;           // 36
constexpr float THRESH = 0.1f;

// ---------------- inverse norms for history + query rows ----------------
__global__ __launch_bounds__(256) void norms_kernel(
    const float* __restrict__ q, const float* __restrict__ hist,
    float* __restrict__ hinv, float* __restrict__ qinv) {
  int row  = blockIdx.x * 8 + (threadIdx.x >> 5);
  int lane = threadIdx.x & 31;
  const float* src;
  if (row < THWn)            src = hist + (size_t)row * Dn;
  else if (row < THWn + HWn) src = q + (size_t)(row - THWn) * Dn;
  else return;
  float ss = 0.f;
  for (int i = lane; i < Dn; i += 32) { float x = src[i]; ss += x * x; }
  for (int m = 16; m >= 1; m >>= 1) ss += __shfl_xor(ss, m, 32);
  if (lane == 0) {
    float inv = 1.0f / fmaxf(sqrtf(ss), 1e-12f);
    if (row < THWn) hinv[row] = inv; else qinv[row - THWn] = inv;
  }
}

// ---------------- M1 = wq^T @ wk   [256 x 256] ----------------
__global__ void m1_kernel(const float* __restrict__ wq, const float* __restrict__ wk,
                          float* __restrict__ M1) {
  int r = blockIdx.x, c = threadIdx.x;
  float acc = 0.f;
  for (int a = 0; a < An; ++a) acc += wq[a * Dn + r] * wk[a * Dn + c];
  M1[r * Dn + c] = acc;
}

// ---------------- QK = q @ M1   [2304 x 256] ----------------
__global__ void qk_kernel(const float* __restrict__ q, const float* __restrict__ M1,
                          float* __restrict__ QK) {
  int n = blockIdx.x, c = threadIdx.x;
  float acc = 0.f;
  for (int d = 0; d < Dn; ++d) acc += q[n * Dn + d] * M1[d * Dn + c];
  QK[n * Dn + c] = acc;
}

// ------- sim GEMM (WMMA f32, 64 q-rows x key-split) fused with streaming top-k -------
__global__ __launch_bounds__(256) void simtopk_kernel(
    const float* __restrict__ q, const float* __restrict__ hist,
    const float* __restrict__ hinv, const float* __restrict__ qinv,
    float* __restrict__ ptval, int* __restrict__ ptidx) {
  __shared__ float qtile[QROWS * Dn];            // 64 KB
  __shared__ float simbuf[NWAVE][QROWS][17];     // padded
  __shared__ float topv[NWAVE][QROWS][17];
  __shared__ int   topi[NWAVE][QROWS][17];

  const int tid    = threadIdx.x;
  const int lane   = tid & 31;
  const int wave   = tid >> 5;
  const int qgroup = blockIdx.x / KEYSPLIT;
  const int ksplit = blockIdx.x % KEYSPLIT;
  const int qbase  = qgroup * QROWS;
  const int tile0  = ksplit * TILES_PER_SPLIT;

  // stage 64 query rows into LDS (vectorized)
  {
    const float4* src = (const float4*)(q + (size_t)qbase * Dn);
    float4* dst = (float4*)qtile;
    for (int i = tid; i < QROWS * Dn / 4; i += 256) dst[i] = src[i];
  }
  // init per-wave top-k lists
  if (lane < 16) {
    for (int r = 0; r < QROWS; ++r) {
      topv[wave][r][lane] = -3.0e38f;
      topi[wave][r][lane] = 0;
    }
  }
  __syncthreads();

  // two rows per lane for top-k maintenance
  const int rowA = lane;
  const int rowB = lane + 32;
  const float qinvA = qinv[qbase + rowA];
  const float qinvB = qinv[qbase + rowB];
  float minv0 = -3.0e38f; int minpos0 = 0;
  float minv1 = -3.0e38f; int minpos1 = 0;

  // WMMA A-fragment addressing (16x4 f32): lanes 0-15 -> K 0,1 ; lanes 16-31 -> K 2,3
  const int arow0 = lane & 15;
  const int kkoff = (lane >> 4) * 2;

  for (int tile = tile0 + wave; tile < tile0 + TILES_PER_SPLIT; tile += NWAVE) {
    const int key0 = tile * 16;
    v8f acc0 = {}, acc1 = {}, acc2 = {}, acc3 = {};
    const float* hbase = hist  + (size_t)(key0 + arow0) * Dn + kkoff; // B: h[N][K]
    const float* qb0   = qtile + (arow0)      * Dn + kkoff;
    const float* qb1   = qtile + (arow0 + 16) * Dn + kkoff;
    const float* qb2   = qtile + (arow0 + 32) * Dn + kkoff;
    const float* qb3   = qtile + (arow0 + 48) * Dn + kkoff;
#pragma unroll
    for (int s = 0; s < Dn / 4; ++s) {
      v2f b  = *(const v2f*)(hbase + s * 4);
      v2f a0 = *(const v2f*)(qb0   + s * 4);
      v2f a1 = *(const v2f*)(qb1   + s * 4);
      v2f a2 = *(const v2f*)(qb2   + s * 4);
      v2f a3 = *(const v2f*)(qb3   + s * 4);
      acc0 = __builtin_amdgcn_wmma_f32_16x16x4_f32(false, a0, false, b, (short)0, acc0, false, false);
      acc1 = __builtin_amdgcn_wmma_f32_16x16x4_f32(false, a1, false, b, (short)0, acc1, false, false);
      acc2 = __builtin_amdgcn_wmma_f32_16x16x4_f32(false, a2, false, b, (short)0, acc2, false, false);
      acc3 = __builtin_amdgcn_wmma_f32_16x16x4_f32(false, a3, false, b, (short)0, acc3, false, false);
    }

    // scatter 16x16 D tiles (raw dots) to LDS: lane = key col, VGPR v = row
#pragma unroll
    for (int v = 0; v < 8; ++v) {
      int r0 = v + 8 * (lane >> 4);
      simbuf[wave][r0][arow0]      = acc0[v];
      simbuf[wave][r0 + 16][arow0] = acc1[v];
      simbuf[wave][r0 + 32][arow0] = acc2[v];
      simbuf[wave][r0 + 48][arow0] = acc3[v];
    }
    __builtin_amdgcn_wave_barrier();

    // each lane scans the 16 candidates of its two rows (replace-min top-k)
    const float* srowA = &simbuf[wave][rowA][0];
    const float* srowB = &simbuf[wave][rowB][0];
#pragma unroll 1
    for (int c = 0; c < 16; ++c) {
      float hv = hinv[key0 + c];
      float vA = srowA[c] * hv * qinvA;
      if (vA > minv0) {
        topv[wave][rowA][minpos0] = vA;
        topi[wave][rowA][minpos0] = key0 + c;
        float mv = 3.0e38f; int mp = 0;
#pragma unroll
        for (int e = 0; e < 16; ++e) {
          float tv = topv[wave][rowA][e];
          if (tv < mv) { mv = tv; mp = e; }
        }
        minv0 = mv; minpos0 = mp;
      }
      float vB = srowB[c] * hv * qinvB;
      if (vB > minv1) {
        topv[wave][rowB][minpos1] = vB;
        topi[wave][rowB][minpos1] = key0 + c;
        float mv = 3.0e38f; int mp = 0;
#pragma unroll
        for (int e = 0; e < 16; ++e) {
          float tv = topv[wave][rowB][e];
          if (tv < mv) { mv = tv; mp = e; }
        }
        minv1 = mv; minpos1 = mp;
      }
    }
    __builtin_amdgcn_wave_barrier();
  }

  __syncthreads();

  // merge 8 per-wave lists (128 candidates) -> per-(row, keysplit) top-16 partial
  if (tid < QROWS) {
    const int r = tid;
    for (int slot = 0; slot < Kn; ++slot) {
      float best = -3.5e38f; int bw = 0, be = 0;
      for (int w = 0; w < NWAVE; ++w) {
        for (int e = 0; e < 16; ++e) {
          float v = topv[w][r][e];
          if (v > best) { best = v; bw = w; be = e; }
        }
      }
      ptval[((size_t)(qbase + r) * KEYSPLIT + ksplit) * Kn + slot] = best;
      ptidx[((size_t)(qbase + r) * KEYSPLIT + ksplit) * Kn + slot] = topi[bw][r][be];
      topv[bw][r][be] = -3.6e38f;
    }
  }
}

// ------- final merge of KEYSPLIT partial top-16 lists + temporal mask -------
__global__ __launch_bounds__(256) void merge_kernel(
    const float* __restrict__ ptval, const int* __restrict__ ptidx,
    float* __restrict__ tval, int* __restrict__ tidx) {
  constexpr int C = KEYSPLIT * Kn;   // 128 candidates per row
  __shared__ float cv[64][C + 1];    // padded vs bank conflicts
  __shared__ int   ci[64][C + 1];
  const int rbase = blockIdx.x * 64;
  for (int i = threadIdx.x; i < 64 * C; i += 256) {
    int r = i >> 7, j = i & (C - 1);
    cv[r][j] = ptval[(size_t)rbase * C + i];
    ci[r][j] = ptidx[(size_t)rbase * C + i];
  }
  __syncthreads();
  if (threadIdx.x < 64) {
    const int r = threadIdx.x;
    for (int slot = 0; slot < Kn; ++slot) {
      float best = -3.5e38f; int bj = 0;
      for (int j = 0; j < C; ++j) {
        float v = cv[r][j];
        if (v > best) { best = v; bj = j; }
      }
      int key = ci[r][bj];
      cv[r][bj] = -3.6e38f;
      int kt = key / HWn;                         // key time index
      tval[(rbase + r) * Kn + slot] = (kt == Tn - 1) ? -1.0f : best;
      tidx[(rbase + r) * Kn + slot] = key;
    }
  }
}

// ---------------- per-query attention over the 16 selected keys ----------------
__global__ __launch_bounds__(256) void attn_kernel(
    const float* __restrict__ hist, const float* __restrict__ QK,
    const float* __restrict__ tval, const int* __restrict__ tidx,
    const float* __restrict__ wv, const float* __restrict__ wo,
    float* __restrict__ out) {
  __shared__ float attnS[8][16];
  __shared__ int   idxS[8][16];
  __shared__ float gbar[8][Dn];
  __shared__ float yS[8][An];

  const int lane = threadIdx.x & 31;
  const int w    = threadIdx.x >> 5;
  const int n    = blockIdx.x * 8 + w;

  if (lane < 16) idxS[w][lane] = tidx[n * Kn + lane];
  __syncthreads();

  // raw scores: QK[n] . hist[idx_k]  (scaled by 1/sqrt(A))
  const float* qrow = QK + (size_t)n * Dn;
  for (int k = 0; k < 16; ++k) {
    const float* hrow = hist + (size_t)idxS[w][k] * Dn;
    float p = 0.f;
#pragma unroll
    for (int i = 0; i < 8; ++i) { int d = lane + 32 * i; p += qrow[d] * hrow[d]; }
    for (int m = 16; m >= 1; m >>= 1) p += __shfl_xor(p, m, 32);
    if (lane == 0) attnS[w][k] = p * 0.08838834764831845f;
  }
  __syncthreads();

  // masked softmax over 16 (lanes 16-31 act as -inf fillers)
  bool valid = (lane < 16) && (tval[n * Kn + (lane & 15)] > THRESH);
  float s = valid ? attnS[w][lane & 15] : -3.0e38f;
  float mx = s;
  for (int m = 16; m >= 1; m >>= 1) mx = fmaxf(mx, __shfl_xor(mx, m, 32));
  float e = valid ? expf(s - mx) : 0.f;
  float se = e;
  for (int m = 16; m >= 1; m >>= 1) se += __shfl_xor(se, m, 32);
  float ak = (se > 0.f) ? e / se : 0.f;   // all-masked row -> zeros (valid_rows)
  __syncthreads();
  if (lane < 16) attnS[w][lane] = ak;
  __syncthreads();

  // gbar = sum_k attn_k * gathered_k   (256-vector)
#pragma unroll
  for (int i = 0; i < 8; ++i) {
    int d = lane + 32 * i;
    float g = 0.f;
    for (int k = 0; k < 16; ++k)
      g += attnS[w][k] * hist[(size_t)idxS[w][k] * Dn + d];
    gbar[w][d] = g;
  }
  __syncthreads();

  // y = wv @ gbar   (128)
#pragma unroll
  for (int j = 0; j < 4; ++j) {
    int a = lane + 32 * j;
    const float* wrow = wv + (size_t)a * Dn;
    float acc = 0.f;
    for (int d = 0; d < Dn; ++d) acc += wrow[d] * gbar[w][d];
    yS[w][a] = acc;
  }
  __syncthreads();

  // out = wo @ y   (256)
#pragma unroll
  for (int i = 0; i < 8; ++i) {
    int d = lane + 32 * i;
    const float* worow = wo + (size_t)d * An;
    float acc = 0.f;
    for (int a = 0; a < An; ++a) acc += worow[a] * yS[w][a];
    out[(size_t)n * Dn + d] = acc;
  }
}

extern "C" void kernel_launch(void* const* d_in, const int* in_sizes, int n_in,
                              void* d_out, int out_size, void* d_ws, size_t ws_size,
                              hipStream_t stream) {
  const float* q    = (const float*)d_in[0];   // [48,48,256]
  const float* hist = (const float*)d_in[1];   // [32,48,48,256]
  const float* wq   = (const float*)d_in[2];   // [128,256]
  const float* wk   = (const float*)d_in[3];   // [128,256]
  const float* wv   = (const float*)d_in[4];   // [128,256]
  const float* wo   = (const float*)d_in[5];   // [256,128]
  float* out = (float*)d_out;

  float* ws    = (float*)d_ws;
  float* hinv  = ws;                                   // 73728
  float* qinv  = hinv + THWn;                          // 2304
  float* M1    = qinv + HWn;                           // 65536
  float* QK    = M1 + Dn * Dn;                         // 589824
  float* ptval = QK + (size_t)HWn * Dn;                // 2304*8*16 = 294912
  int*   ptidx = (int*)(ptval + (size_t)HWn * KEYSPLIT * Kn);   // 294912
  float* tval  = (float*)(ptidx + (size_t)HWn * KEYSPLIT * Kn); // 36864
  int*   tidx  = (int*)(tval + (size_t)HWn * Kn);               // 36864

  norms_kernel<<<(THWn + HWn) / 8, 256, 0, stream>>>(q, hist, hinv, qinv);
  m1_kernel<<<Dn, Dn, 0, stream>>>(wq, wk, M1);
  qk_kernel<<<HWn, Dn, 0, stream>>>(q, M1, QK);
  simtopk_kernel<<<QGROUPS * KEYSPLIT, 256, 0, stream>>>(q, hist, hinv, qinv, ptval, ptidx);
  merge_kernel<<<HWn / 64, 256, 0, stream>>>(ptval, ptidx, tval, tidx);
  attn_kernel<<<HWn / 8, 256, 0, stream>>>(hist, QK, tval, tidx, wv, wo, out);
}